// QuantumStateEncoder_54417235640460
// MI455X (gfx1250) — compile-verified
//
#include <hip/hip_runtime.h>
#include <math.h>

typedef __attribute__((ext_vector_type(2))) float v2f;
typedef __attribute__((ext_vector_type(8))) float v8f;

#define NQ 8
#define NSTATES 256
#define WPB 8                 // waves (batch elements) per block
#define BLOCK (WPB * 32)

// tanh via native exp: monotone, saturates correctly at +-1, no NaN at large |x|
__device__ __forceinline__ float tanh_fast(float x) {
    float e = __expf(2.0f * x);
    return 1.0f - 2.0f / (e + 1.0f);
}

// CZ diagonal sign for basis index i (8 qubits, nearest-neighbor pairs 0..6)
__device__ __forceinline__ float czsign(int i) {
    int p = __popc(i & (i >> 1) & 0x7F);
    return (p & 1) ? -1.0f : 1.0f;
}

// Ry 2x2 entry: [[c,-s],[s,c]][i][j]
__device__ __forceinline__ float rent(float c, float s, int i, int j) {
    float off = (i == 1) ? s : -s;
    return (i == j) ? c : off;
}

// 4-qubit tensor-product matrix entry: prod_q R_q[row_q, col_q]
__device__ __forceinline__ float entry4(const float* cc, const float* ss, int row, int col) {
    float p = 1.0f;
#pragma unroll
    for (int q = 0; q < 4; ++q)
        p *= rent(cc[q], ss[q], (row >> q) & 1, (col >> q) & 1);
    return p;
}

__global__ __launch_bounds__(BLOCK)
void qenc_kernel(const float* __restrict__ pf,     // (nbatch, 8)
                 const float* __restrict__ theta,  // (2, 8)
                 const float* __restrict__ scale,  // (8,)
                 float* __restrict__ out,          // (nbatch, 256)
                 int nbatch) {
    // All LDS traffic below is wave-private (indexed by this wave's slot only),
    // and CDNA5 LDS ops from one wave are processed in order, so no workgroup
    // barriers are needed -- only compiler scheduling barriers.
    __shared__ float cs[WPB * 32];        // per-slot: c0[8] s0[8] c1[8] s1[8]
    __shared__ float ldsT[WPB * NSTATES]; // intermediate T relayout

    const int t    = threadIdx.x;
    const int slot = t >> 5;
    const int lane = t & 31;
    int b = blockIdx.x * WPB + slot;
    if (b >= nbatch) b = nbatch - 1;      // clamp: duplicate work, keeps EXEC full for WMMA

    // ---- cooperative angle -> cos/sin table (1 value per lane, wave-private) ----
    {
        int layer = (lane >> 4) & 1;
        int part  = (lane >> 3) & 1;  // 0 = cos, 1 = sin
        int q     = lane & 7;
        float x   = pf[(size_t)b * NQ + q];
        float ang = tanh_fast(x) * 3.14159265358979323846f * scale[q] + theta[layer * NQ + q];
        float hv  = 0.5f * ang;
        cs[slot * 32 + lane] = part ? __sinf(hv) : __cosf(hv);
    }
    __builtin_amdgcn_wave_barrier();      // keep ds_load after ds_store; LDS is in-order per wave

    // broadcast-read the 32 scalars into registers (8x ds_load_b128, broadcast)
    float arr[32];
#pragma unroll
    for (int i = 0; i < 8; ++i) {
        float4 v4 = *reinterpret_cast<const float4*>(&cs[slot * 32 + 4 * i]);
        arr[4 * i + 0] = v4.x; arr[4 * i + 1] = v4.y;
        arr[4 * i + 2] = v4.z; arr[4 * i + 3] = v4.w;
    }
    const float* c0lo = arr + 0;  const float* c0hi = arr + 4;
    const float* s0lo = arr + 8;  const float* s0hi = arr + 12;
    const float* c1lo = arr + 16; const float* c1hi = arr + 20;
    const float* s1lo = arr + 24; const float* s1hi = arr + 28;

    const int lm = lane & 15;   // row (A) or column (B/D) index
    const int hi = lane >> 4;   // half-wave select

    // layer-1 product-state factor over the high qubits, row = lm
    float u = 1.0f;
#pragma unroll
    for (int q = 0; q < 4; ++q)
        u *= ((lm >> q) & 1) ? s0hi[q] : c0hi[q];

    // ---- matmul 1: T = S1 * L^T  (A = S1, B[k][n] = L[n][k], both built in-layout) ----
    v8f tacc = {0.f, 0.f, 0.f, 0.f, 0.f, 0.f, 0.f, 0.f};
#pragma unroll
    for (int kc = 0; kc < 4; ++kc) {
        int k0 = 4 * kc + 2 * hi;
        v2f a, bm;
#pragma unroll
        for (int d = 0; d < 2; ++d) {
            int k = k0 + d;
            float v = 1.0f;
#pragma unroll
            for (int q = 0; q < 4; ++q)
                v *= ((k >> q) & 1) ? s0lo[q] : c0lo[q];
            float s1e = u * v * czsign(lm * 16 + k);        // A[lm][k] = S1[lm][k]
            float bte = entry4(c1lo, s1lo, lm, k);          // B[k][lm] = L[lm][k]
            if (d == 0) { a.x = s1e; bm.x = bte; } else { a.y = s1e; bm.y = bte; }
        }
        tacc = __builtin_amdgcn_wmma_f32_16x16x4_f32(false, a, false, bm,
                                                     (short)0, tacc, false, false);
    }

    // ---- relayout T (D layout -> B layout) through wave-private LDS ----
#pragma unroll
    for (int r = 0; r < 8; ++r)
        ldsT[slot * NSTATES + (r + 8 * hi) * 16 + lm] = tacc[r];
    __builtin_amdgcn_wave_barrier();      // same-wave LDS in-order; just block reordering

    // ---- matmul 2: S2 = H * T ----
    v8f acc = {0.f, 0.f, 0.f, 0.f, 0.f, 0.f, 0.f, 0.f};
#pragma unroll
    for (int kc = 0; kc < 4; ++kc) {
        int k0 = 4 * kc + 2 * hi;
        v2f a, bm;
        a.x  = entry4(c1hi, s1hi, lm, k0);                  // A[lm][k] = H[lm][k]
        a.y  = entry4(c1hi, s1hi, lm, k0 + 1);
        bm.x = ldsT[slot * NSTATES + (k0    ) * 16 + lm];   // B[k][lm] = T[k][lm]
        bm.y = ldsT[slot * NSTATES + (k0 + 1) * 16 + lm];
        acc = __builtin_amdgcn_wmma_f32_16x16x4_f32(false, a, false, bm,
                                                    (short)0, acc, false, false);
    }

    // ---- CZ phase, norm, store ----
    float res[8];
    float sumsq = 0.0f;
#pragma unroll
    for (int r = 0; r < 8; ++r) {
        int m = r + 8 * hi;
        float v = acc[r] * czsign(m * 16 + lm);
        res[r] = v;
        sumsq += v * v;
    }
#pragma unroll
    for (int off = 16; off > 0; off >>= 1)
        sumsq += __shfl_xor(sumsq, off, 32);
    float inv = 1.0f / fmaxf(sqrtf(sumsq), 1e-12f);

    size_t base = (size_t)b * NSTATES;
#pragma unroll
    for (int r = 0; r < 8; ++r) {
        int m = r + 8 * hi;
        out[base + (size_t)(m * 16 + lm)] = res[r] * inv;
    }
}

extern "C" void kernel_launch(void* const* d_in, const int* in_sizes, int n_in,
                              void* d_out, int out_size, void* d_ws, size_t ws_size,
                              hipStream_t stream) {
    const float* pf    = (const float*)d_in[0]; // (nbatch, 8)
    const float* theta = (const float*)d_in[1]; // (2, 8)
    const float* scale = (const float*)d_in[2]; // (8,)
    float* out = (float*)d_out;

    int nbatch = in_sizes[0] / NQ;              // 65536
    int blocks = (nbatch + WPB - 1) / WPB;      // 8192
    qenc_kernel<<<blocks, BLOCK, 0, stream>>>(pf, theta, scale, out, nbatch);
}